// DeepseekV3MoETTNN_71803263255220
// MI455X (gfx1250) — compile-verified
//
#include <hip/hip_runtime.h>
#include <hip/hip_bf16.h>

// ---------------- problem constants (DeepSeek-V3 noaux_tc routing) ----------
#define T_TOK   256
#define H_DIM   1024
#define E_EXP   256
#define I_DIM   256
#define N_GROUP 8
#define TOPK_G  4
#define TOP_K   8
#define SCALE_F 2.5f

typedef __attribute__((ext_vector_type(2))) float v2f;
typedef __attribute__((ext_vector_type(8))) float v8f;

#define WMMA_F32(a, b, c) \
  __builtin_amdgcn_wmma_f32_16x16x4_f32(false, (a), false, (b), (short)0, (c), false, false)

__device__ __forceinline__ float fast_silu(float v) {
  return __fdividef(v, 1.0f + __expf(-v));
}
__device__ __forceinline__ float fast_sigmoid(float v) {
  return __fdividef(1.0f, 1.0f + __expf(-v));
}

// ---------------- shared LDS tile layout for the MLP kernels ----------------
struct MlpSmem {
  float x[16][H_DIM + 4];   // 16 token rows of hidden state (padded vs bank conflicts)
  float h1[16][I_DIM + 4];  // silu(x*Wg) .* (x*Wu)
  int   toks[16];
  float wts[16];
};

// One 16-token tile through one expert MLP:  out[tok] (+)= wt * (silu(xWg).*(xWu)) Wd
// 256 threads = 8 wave32s. Stage A: each wave owns 32 cols of I (2 N-tiles, gate+up).
// Stage B: each wave owns 8 N-tiles of H (interleaved).
__device__ __forceinline__ void mlp_tile(MlpSmem& sm, int count,
                                         const float* __restrict__ Wg,
                                         const float* __restrict__ Wu,
                                         const float* __restrict__ Wd,
                                         float* __restrict__ out,
                                         bool accumulate) {
  const int tid  = threadIdx.x;
  const int wave = tid >> 5;
  const int lane = tid & 31;
  const int half = lane >> 4;
  const int l16  = lane & 15;

  // ---------------- stage A: h1 = silu(X @ Wg) * (X @ Wu) ------------------
  v8f cg0 = {}, cg1 = {}, cu0 = {}, cu1 = {};
  const int nbase = wave * 32;  // this wave's 32 columns of I
  for (int k = 0; k < H_DIM; k += 4) {
    const int kr = k + 2 * half;
    v2f a;
    a.x = sm.x[l16][kr];
    a.y = sm.x[l16][kr + 1];
    const float* g0 = Wg + (size_t)kr * I_DIM + nbase + l16;
    const float* u0 = Wu + (size_t)kr * I_DIM + nbase + l16;
    v2f bg0, bg1, bu0, bu1;
    bg0.x = __builtin_nontemporal_load(g0);
    bg0.y = __builtin_nontemporal_load(g0 + I_DIM);
    bg1.x = __builtin_nontemporal_load(g0 + 16);
    bg1.y = __builtin_nontemporal_load(g0 + I_DIM + 16);
    bu0.x = __builtin_nontemporal_load(u0);
    bu0.y = __builtin_nontemporal_load(u0 + I_DIM);
    bu1.x = __builtin_nontemporal_load(u0 + 16);
    bu1.y = __builtin_nontemporal_load(u0 + I_DIM + 16);
    cg0 = WMMA_F32(a, bg0, cg0);
    cg1 = WMMA_F32(a, bg1, cg1);
    cu0 = WMMA_F32(a, bu0, cu0);
    cu1 = WMMA_F32(a, bu1, cu1);
  }
#pragma unroll
  for (int r = 0; r < 8; ++r) {
    const int m = r + 8 * half;  // C layout: VGPR r -> row r (lanes<16) / r+8 (lanes>=16)
    sm.h1[m][nbase + l16]      = fast_silu(cg0[r]) * cu0[r];
    sm.h1[m][nbase + 16 + l16] = fast_silu(cg1[r]) * cu1[r];
  }
  __syncthreads();

  // ---------------- stage B: out (+)= (h1 @ Wd) * wt -----------------------
  v8f cd[8] = {};
  for (int k = 0; k < I_DIM; k += 4) {
    const int kr = k + 2 * half;
    v2f a;
    a.x = sm.h1[l16][kr];
    a.y = sm.h1[l16][kr + 1];
#pragma unroll
    for (int j = 0; j < 8; ++j) {
      const float* d0 = Wd + (size_t)kr * H_DIM + (wave + 8 * j) * 16 + l16;
      v2f b;
      b.x = __builtin_nontemporal_load(d0);
      b.y = __builtin_nontemporal_load(d0 + H_DIM);
      cd[j] = WMMA_F32(a, b, cd[j]);
    }
  }
#pragma unroll
  for (int j = 0; j < 8; ++j) {
    const int n = (wave + 8 * j) * 16 + l16;
#pragma unroll
    for (int r = 0; r < 8; ++r) {
      const int m = r + 8 * half;
      if (m < count) {
        const float v = cd[j][r] * sm.wts[m];
        float* p = out + (size_t)sm.toks[m] * H_DIM + n;
        if (accumulate) atomicAdd(p, v);
        else            *p = v;
      }
    }
  }
}

__device__ __forceinline__ void load_x_tile_lds(float (*dst)[H_DIM + 4],
                                                const int* toks,
                                                const float* __restrict__ x) {
  // 16 rows x H floats; 256 threads, float4 each
  for (int i = threadIdx.x; i < 16 * (H_DIM / 4); i += 256) {
    const int m = i >> 8;             // i / (H/4)
    const int c = (i & 255) << 2;     // (i % (H/4)) * 4
    const float4 v = *(const float4*)(x + (size_t)toks[m] * H_DIM + c);
    dst[m][c]     = v.x;
    dst[m][c + 1] = v.y;
    dst[m][c + 2] = v.z;
    dst[m][c + 3] = v.w;
  }
}

// ---------------- kernel 0: zero per-expert counters ------------------------
__global__ void init_kernel(int* expert_count) {
  if (threadIdx.x < E_EXP) expert_count[threadIdx.x] = 0;
}

// ---------------- kernel 1: WMMA gate + grouped top-k routing ---------------
// 16 blocks x 16 tokens. logits[16,E] = X_tile @ gate_w^T via f32 WMMA.
// gate_w is [E,H] row-major, so B[k][n] = gw[n*H+k]: k,k+1 contiguous -> b64.
__global__ void __launch_bounds__(256)
gate_kernel(const float* __restrict__ x, const float* __restrict__ gw,
            const float* __restrict__ eb, int* expert_count,
            int* expert_tokens, float* expert_wts) {
  __shared__ float xs[16][H_DIM + 4];     // ~64 KB
  __shared__ float scores[16][E_EXP];     // raw sigmoid scores, 16 KB
  __shared__ float sc[16][E_EXP + 1];     // + correction bias (padded rows)
  __shared__ int   tk[16];

  const int t0 = blockIdx.x * 16;
  if (threadIdx.x < 16) tk[threadIdx.x] = t0 + threadIdx.x;
  __syncthreads();
  load_x_tile_lds(xs, tk, x);
  __syncthreads();

  const int wave = threadIdx.x >> 5;
  const int lane = threadIdx.x & 31;
  const int half = lane >> 4;
  const int l16  = lane & 15;
  const int n0   = wave * 32;  // this wave's 32 experts

  v8f c0 = {}, c1 = {};
  const float* p0base = gw + (size_t)(n0 + l16) * H_DIM;
  const float* p1base = gw + (size_t)(n0 + 16 + l16) * H_DIM;
  for (int k = 0; k < H_DIM; k += 4) {
    const int kr = k + 2 * half;
    v2f a;
    a.x = xs[l16][kr];
    a.y = xs[l16][kr + 1];
    const v2f b0 = *(const v2f*)(p0base + kr);  // global_load_b64
    const v2f b1 = *(const v2f*)(p1base + kr);
    c0 = WMMA_F32(a, b0, c0);
    c1 = WMMA_F32(a, b1, c1);
  }
#pragma unroll
  for (int r = 0; r < 8; ++r) {
    const int m = r + 8 * half;
    const int n = n0 + l16;
    const float s0 = fast_sigmoid(c0[r]);
    const float s1 = fast_sigmoid(c1[r]);
    scores[m][n]      = s0;
    scores[m][n + 16] = s1;
    sc[m][n]      = s0 + eb[n];
    sc[m][n + 16] = s1 + eb[n + 16];
  }
  __syncthreads();

  // ---- routing: one token per thread (16 active), serial per token ----
  if (threadIdx.x < 16) {
    const int m = threadIdx.x;
    const int t = t0 + m;
    float gs[N_GROUP];
#pragma unroll
    for (int g = 0; g < N_GROUP; ++g) {  // top-2-per-group sums
      float m1 = -3.4e38f, m2 = -3.4e38f;
      for (int i = 0; i < E_EXP / N_GROUP; ++i) {
        const float v = sc[m][g * (E_EXP / N_GROUP) + i];
        if (v > m1)      { m2 = m1; m1 = v; }
        else if (v > m2) { m2 = v; }
      }
      gs[g] = m1 + m2;
    }
    unsigned gmask = 0u;
    for (int it = 0; it < TOPK_G; ++it) {  // top-4 groups (lowest index on tie)
      float best = -3.4e38f; int bi = 0;
      for (int g = 0; g < N_GROUP; ++g)
        if (!((gmask >> g) & 1u) && gs[g] > best) { best = gs[g]; bi = g; }
      gmask |= 1u << bi;
    }
    int   idx[TOP_K];
    float wts[TOP_K];
    float sum = 0.f;
    for (int kk = 0; kk < TOP_K; ++kk) {  // top-8 experts within selected groups
      float best = -3.4e38f; int bi = 0;
      for (int e = 0; e < E_EXP; ++e) {
        if (!((gmask >> (e >> 5)) & 1u)) continue;
        const float v = sc[m][e];
        if (v > best) { best = v; bi = e; }
      }
      idx[kk] = bi;
      wts[kk] = scores[m][bi];  // weights come from RAW sigmoid scores
      sum += scores[m][bi];
      sc[m][bi] = -3.4e38f;
    }
    const float inv = SCALE_F / (sum + 1e-20f);
#pragma unroll
    for (int kk = 0; kk < TOP_K; ++kk) {
      const int e = idx[kk];
      const int pos = atomicAdd(&expert_count[e], 1);
      expert_tokens[e * T_TOK + pos] = t;
      expert_wts[e * T_TOK + pos]    = wts[kk] * inv;
    }
  }
}

// ---------------- kernel 2: shared expert (writes/initializes d_out) -------
__global__ void __launch_bounds__(256)
shared_kernel(const float* __restrict__ x, const float* __restrict__ swg,
              const float* __restrict__ swu, const float* __restrict__ swd,
              float* __restrict__ out) {
  __shared__ MlpSmem sm;
  const int t0 = blockIdx.x * 16;
  if (threadIdx.x < 16) {
    sm.toks[threadIdx.x] = t0 + threadIdx.x;
    sm.wts[threadIdx.x]  = 1.0f;
  }
  __syncthreads();
  load_x_tile_lds(sm.x, sm.toks, x);
  __syncthreads();
  mlp_tile(sm, 16, swg, swu, swd, out, /*accumulate=*/false);
}

// ---------------- kernel 3: routed experts (atomic-add into d_out) ---------
__global__ void __launch_bounds__(256)
expert_kernel(const float* __restrict__ x, const float* __restrict__ wg,
              const float* __restrict__ wu, const float* __restrict__ wd,
              const int* __restrict__ expert_count,
              const int* __restrict__ expert_tokens,
              const float* __restrict__ expert_wts,
              float* __restrict__ out) {
  __shared__ MlpSmem sm;
  const int e       = blockIdx.x >> 4;   // 16 chunks of 16 tokens per expert
  const int chunk   = blockIdx.x & 15;
  const int cnt_all = expert_count[e];
  const int start   = chunk * 16;
  if (start >= cnt_all) return;          // uniform per block
  const int count = (cnt_all - start < 16) ? (cnt_all - start) : 16;

  if (threadIdx.x < 16) {
    const int m = threadIdx.x;
    int tok = 0; float w = 0.f;
    if (m < count) {
      tok = expert_tokens[e * T_TOK + start + m];
      w   = expert_wts[e * T_TOK + start + m];
    }
    sm.toks[m] = tok;  // padded rows point at token 0 (finite data, weight 0, store skipped)
    sm.wts[m]  = w;
  }
  __syncthreads();
  load_x_tile_lds(sm.x, sm.toks, x);
  __syncthreads();
  mlp_tile(sm, count,
           wg + (size_t)e * H_DIM * I_DIM,
           wu + (size_t)e * H_DIM * I_DIM,
           wd + (size_t)e * I_DIM * H_DIM,
           out, /*accumulate=*/true);
}

// ---------------- host-side launch ------------------------------------------
extern "C" void kernel_launch(void* const* d_in, const int* in_sizes, int n_in,
                              void* d_out, int out_size, void* d_ws, size_t ws_size,
                              hipStream_t stream) {
  (void)in_sizes; (void)n_in; (void)out_size; (void)ws_size;
  const float* x   = (const float*)d_in[0];  // [1,1,T,H]
  const float* gw  = (const float*)d_in[1];  // [E,H]
  const float* eb  = (const float*)d_in[2];  // [E]
  const float* wg  = (const float*)d_in[3];  // [E,H,I]
  const float* wu  = (const float*)d_in[4];  // [E,H,I]
  const float* wdn = (const float*)d_in[5];  // [E,I,H]
  const float* swg = (const float*)d_in[6];  // [H,I]
  const float* swu = (const float*)d_in[7];  // [H,I]
  const float* swd = (const float*)d_in[8];  // [I,H]
  float* out = (float*)d_out;

  char* ws = (char*)d_ws;
  int*   expert_count  = (int*)ws;                                   // E ints
  int*   expert_tokens = (int*)(ws + 1024);                          // E*T ints
  float* expert_wts    = (float*)(ws + 1024 + E_EXP * T_TOK * 4);    // E*T floats

  init_kernel<<<1, 256, 0, stream>>>(expert_count);
  gate_kernel<<<T_TOK / 16, 256, 0, stream>>>(x, gw, eb, expert_count,
                                              expert_tokens, expert_wts);
  shared_kernel<<<T_TOK / 16, 256, 0, stream>>>(x, swg, swu, swd, out);
  expert_kernel<<<E_EXP * 16, 256, 0, stream>>>(x, wg, wu, wdn, expert_count,
                                                expert_tokens, expert_wts, out);
}